// LSTMHead_81758997447283
// MI455X (gfx1250) — compile-verified
//
#include <hip/hip_runtime.h>

typedef float v2f __attribute__((ext_vector_type(2)));
typedef float v4f __attribute__((ext_vector_type(4)));
typedef float v8f __attribute__((ext_vector_type(8)));

#define SEQ   32768
#define DIM   1024
#define ROWS_PER_WAVE 16
#define WAVES_PER_BLOCK 8

// -------------------------------------------------------------------------
// Kernel 1: gates[s][g] = dot(x[s,:], W_ih[g,:]) + b_ih[g] + b_hh[g]
// One wave computes 16 rows via V_WMMA_F32_16X16X4_F32, K-loop of 256 steps.
// A (16x4 f32): lane m<16 holds row m, VGPR0/1 = K {0,1}; lanes 16-31 K {2,3}.
// B (4x16 f32): column n = gate (n&3), replicated across N lanes (branchless,
// replicated columns 4..15 are simply never stored).
// -------------------------------------------------------------------------
__global__ __launch_bounds__(WAVES_PER_BLOCK * 32)
void gates_gemm_wmma(const float* __restrict__ x,
                     const float* __restrict__ Wih,
                     const float* __restrict__ bih,
                     const float* __restrict__ bhh,
                     float* __restrict__ gates)
{
    const int lane    = threadIdx.x & 31;
    const int wave    = threadIdx.x >> 5;
    const int lane_lo = lane & 15;
    const int lane_hi = lane >> 4;           // 0: K pair {k,k+1}, 1: {k+2,k+3}
    const int row_base = (blockIdx.x * WAVES_PER_BLOCK + wave) * ROWS_PER_WAVE;

    // A operand source: my row of x, offset by 2*lane_hi in K
    const float* xrow = x + (size_t)(row_base + lane_lo) * DIM + 2 * lane_hi;
    // B operand source: gate row (lane_lo & 3) of W_ih (16 KB, lives in WGP$)
    const float* wrow = Wih + (size_t)(lane_lo & 3) * DIM + 2 * lane_hi;

    v8f acc = {};
    #pragma unroll 4
    for (int k = 0; k < DIM; k += 4) {
        v2f a = *(const v2f*)(xrow + k);     // global_load_b64
        v2f b = *(const v2f*)(wrow + k);     // hits cache after first pass
        acc = __builtin_amdgcn_wmma_f32_16x16x4_f32(
                  /*neg_a=*/false, a, /*neg_b=*/false, b,
                  /*c_mod=*/(short)0, acc, /*reuse_a=*/false, /*reuse_b=*/false);
    }

    // D layout: lane n (n<16) holds column n, rows 0..7 in acc[0..7];
    // lane n+16 holds column n, rows 8..15. We need columns 0..3 only.
    if (lane_lo < 4) {
        const int g    = lane_lo;
        const float bb = bih[g] + bhh[g];
        const int r0   = row_base + lane_hi * 8;
        #pragma unroll
        for (int r = 0; r < 8; ++r)
            gates[(size_t)(r0 + r) * 4 + g] = acc[r] + bb;
    }
}

// -------------------------------------------------------------------------
// Kernel 2: serial LSTM scan, scalar hidden state. Latency-bound: minimize
// the dependent chain with hardware exp2/rcp TRANS ops.
// -------------------------------------------------------------------------
__device__ __forceinline__ float fsig(float xv)
{
    // sigmoid(x) = 1 / (1 + 2^(-x*log2e))     (v_exp_f32 + v_rcp_f32)
    return __builtin_amdgcn_rcpf(1.0f + __builtin_amdgcn_exp2f(-1.4426950408889634f * xv));
}
__device__ __forceinline__ float ftanh(float xv)
{
    // tanh(x) = 2*sigmoid(2x) - 1
    return fmaf(2.0f,
                __builtin_amdgcn_rcpf(1.0f + __builtin_amdgcn_exp2f(-2.8853901631779268f * xv)),
                -1.0f);
}

__global__ void lstm_scan(const float* __restrict__ gates,
                          const float* __restrict__ Whh,
                          const float* __restrict__ h0,
                          const float* __restrict__ c0,
                          float* __restrict__ out)
{
    if (threadIdx.x != 0) return;

    const float w0 = Whh[0], w1 = Whh[1], w2 = Whh[2], w3 = Whh[3];
    float h = h0[0];
    float c = c0[0];

    v4f hbuf = {};
    for (int t = 0; t < SEQ; ++t) {
        v4f g = *(const v4f*)(gates + 4 * (size_t)t);   // global_load_b128
        if ((t & 31) == 0)
            __builtin_prefetch(gates + 4 * (size_t)t + 512, 0, 3); // global_prefetch_b8

        const float gi = fmaf(w0, h, g.x);
        const float gf = fmaf(w1, h, g.y);
        const float gc = fmaf(w2, h, g.z);
        const float go = fmaf(w3, h, g.w);

        const float iv = fsig(gi);
        const float fv = fsig(gf);
        const float gv = ftanh(gc);
        const float ov = fsig(go);

        c = fmaf(fv, c, iv * gv);
        h = ov * ftanh(c);

        hbuf[t & 3] = h;
        if ((t & 3) == 3)
            *(v4f*)(out + (t - 3)) = hbuf;              // global_store_b128
    }
}

// -------------------------------------------------------------------------
// Launcher. Inputs: x, W_ih, W_hh, b_ih, b_hh, h0, c0 (all fp32).
// d_ws stages the gate matrix (SEQ*4 floats = 512 KB).
// -------------------------------------------------------------------------
extern "C" void kernel_launch(void* const* d_in, const int* in_sizes, int n_in,
                              void* d_out, int out_size, void* d_ws, size_t ws_size,
                              hipStream_t stream)
{
    const float* x    = (const float*)d_in[0];
    const float* Wih  = (const float*)d_in[1];
    const float* Whh  = (const float*)d_in[2];
    const float* bih  = (const float*)d_in[3];
    const float* bhh  = (const float*)d_in[4];
    const float* h0   = (const float*)d_in[5];
    const float* c0   = (const float*)d_in[6];
    float*       out  = (float*)d_out;
    float*       gts  = (float*)d_ws;

    const int rows_per_block = ROWS_PER_WAVE * WAVES_PER_BLOCK;   // 128
    const int nblocks = SEQ / rows_per_block;                     // 256

    gates_gemm_wmma<<<nblocks, WAVES_PER_BLOCK * 32, 0, stream>>>(x, Wih, bih, bhh, gts);
    lstm_scan<<<1, 32, 0, stream>>>(gts, Whh, h0, c0, out);
}